// CapsuleLayer_76244259439239
// MI455X (gfx1250) — compile-verified
//
#include <hip/hip_runtime.h>
#include <math.h>

typedef __attribute__((ext_vector_type(2))) float v2f;
typedef __attribute__((ext_vector_type(8))) float v8f;

#define B_SZ   32
#define N_CAP  64      // both input and output capsules
#define D_CAP  128     // capsule dim == input dim
#define M_ROWS (N_CAP * D_CAP)        // 8192 rows per j-GEMM
#define U_PER_B (N_CAP * M_ROWS)      // 524288 floats per batch in u
#define LDA 132        // padded LDS stride for W/x tiles (4*row spread -> conflict free)
#define LDT 67         // padded LDS stride for transpose tile

// ---------------------------------------------------------------------------
// Kernel A: u[b,j,k,d] = sum_e W[j,k,d,e] * x[b,j,e]
// Per block: one j (blockIdx.y), one 64-row m-tile (blockIdx.x), 128 threads
// (4 waves). Each wave computes a 16(m) x 32(b) tile via chained
// v_wmma_f32_16x16x4_f32 over K=128 (32 steps, 2 n-halves).
// ---------------------------------------------------------------------------
__global__ __launch_bounds__(128)
void caps_gemm_wmma(const float* __restrict__ W, const float* __restrict__ x,
                    float* __restrict__ u) {
    __shared__ float Wlds[64 * LDA];     // 64 rows x 128 e (padded)
    __shared__ float Xlds[32 * LDA];     // 32 b   x 128 e (padded)
    __shared__ float Tr[32 * LDT];       // transpose tile: 32 n x 64 m (padded)

    const int t     = threadIdx.x;
    const int j     = blockIdx.y;
    const int mtile = blockIdx.x;        // 128 tiles of 64 rows

    // ---- stage W tile: 64*128 floats, fully contiguous in global ----
    const float* Wt = W + (size_t)j * (size_t)(M_ROWS * D_CAP)
                        + (size_t)mtile * 64 * D_CAP;
    const float4* Wt4 = (const float4*)Wt;
#pragma unroll
    for (int i = 0; i < 16; ++i) {
        int f4 = i * 128 + t;            // 2048 float4 total
        float4 v = Wt4[f4];
        int r = f4 >> 5;                 // /32 float4 per row
        int e = (f4 & 31) << 2;
        *(float4*)&Wlds[r * LDA + e] = v;
    }
    // ---- stage x slice: x[b, j, e], 32 rows of 128 floats ----
#pragma unroll
    for (int i = 0; i < 8; ++i) {
        int f4 = i * 128 + t;            // 1024 float4 total
        int b  = f4 >> 5;
        int e  = (f4 & 31) << 2;
        float4 v = *(const float4*)&x[((size_t)b * N_CAP + j) * D_CAP + e];
        *(float4*)&Xlds[b * LDA + e] = v;
    }
    __syncthreads();

    const int wave  = t >> 5;
    const int lane  = t & 31;
    const int lrow  = lane & 15;
    const int lhalf = lane >> 4;         // 0 or 1

    v8f acc0 = {};                       // n = 0..15
    v8f acc1 = {};                       // n = 16..31
    const float* Arow = &Wlds[(wave * 16 + lrow) * LDA];
#pragma unroll
    for (int kk = 0; kk < 32; ++kk) {
        const int e0 = 4 * kk + 2 * lhalf;
        v2f a  = *(const v2f*)&Arow[e0];
        v2f b0 = *(const v2f*)&Xlds[lrow * LDA + e0];
        v2f b1 = *(const v2f*)&Xlds[(16 + lrow) * LDA + e0];
        acc0 = __builtin_amdgcn_wmma_f32_16x16x4_f32(false, a, false, b0,
                                                     (short)0, acc0, false, false);
        acc1 = __builtin_amdgcn_wmma_f32_16x16x4_f32(false, a, false, b1,
                                                     (short)0, acc1, false, false);
    }

    // ---- transpose through LDS: Tr[n][m_local], m_local in 0..63 ----
#pragma unroll
    for (int r = 0; r < 8; ++r) {
        int mloc = wave * 16 + r + 8 * lhalf;   // C/D layout: M = r + 8*(lane/16)
        Tr[lrow * LDT + mloc]        = acc0[r]; // n = lrow
        Tr[(16 + lrow) * LDT + mloc] = acc1[r]; // n = 16 + lrow
    }
    __syncthreads();

    // ---- store u[b][j][m]: per n, 64 consecutive floats ----
    {
        int n    = t >> 1;               // 0..31 (b)
        int mloc = (t & 1) * 32;         // 0 or 32
        float* dst = u + (size_t)n * U_PER_B + (size_t)j * M_ROWS
                       + (size_t)mtile * 64 + mloc;
#pragma unroll
        for (int i = 0; i < 8; ++i) {
            float4 v = *(const float4*)&Tr[n * LDT + mloc + i * 4];
            *(float4*)&dst[i * 4] = v;
        }
    }
}

// ---------------------------------------------------------------------------
// Kernel B/D: one routing step.
//   uniform==1: c = 1/64 (softmax of zeros)
//   uniform==0: c[b,j,k,d] = softmax_k( u[b,j,k,d] * vsum[b,k,d] )
//   s[b,k,d] += sum_j c * u   (atomics over 256 partial blocks)
// grid = (8 j-groups, 32 b), block = 128 threads (thread = d column).
// ---------------------------------------------------------------------------
__global__ __launch_bounds__(128)
void caps_routing_step(const float* __restrict__ u, const float* __restrict__ vsum,
                       float* __restrict__ s, int uniform) {
    __shared__ float sp[N_CAP * D_CAP];  // 32 KB partial s, thread-private columns
    const int t  = threadIdx.x;          // d
    const int b  = blockIdx.y;
    const int j0 = blockIdx.x * 8;

    for (int i = t; i < N_CAP * D_CAP; i += 128) sp[i] = 0.0f;  // own column only

    float vs[N_CAP];
    if (!uniform) {
#pragma unroll
        for (int k = 0; k < N_CAP; ++k)
            vs[k] = vsum[((size_t)b * N_CAP + k) * D_CAP + t];
    }

    const float* ub = u + (size_t)b * U_PER_B + (size_t)j0 * M_ROWS;
    for (int jj = 0; jj < 8; ++jj) {
        const float* uj = ub + (size_t)jj * M_ROWS;
        float uk[N_CAP];
#pragma unroll
        for (int k = 0; k < N_CAP; ++k) uk[k] = uj[k * D_CAP + t];

        if (uniform) {
#pragma unroll
            for (int k = 0; k < N_CAP; ++k)
                sp[k * D_CAP + t] += uk[k] * (1.0f / 64.0f);
        } else {
            float m = -3.402823e38f;
#pragma unroll
            for (int k = 0; k < N_CAP; ++k) m = fmaxf(m, uk[k] * vs[k]);
            float se = 0.0f;
#pragma unroll
            for (int k = 0; k < N_CAP; ++k) se += __expf(uk[k] * vs[k] - m);
            float inv = 1.0f / se;
#pragma unroll
            for (int k = 0; k < N_CAP; ++k) {
                float c = __expf(uk[k] * vs[k] - m) * inv;
                sp[k * D_CAP + t] += c * uk[k];
            }
        }
    }
    // flush partials (thread-private columns -> no barrier needed)
    float* sb = s + (size_t)b * (N_CAP * D_CAP);
    for (int i = t; i < N_CAP * D_CAP; i += 128) atomicAdd(&sb[i], sp[i]);
}

// ---------------------------------------------------------------------------
// Kernel C: squash along d, accumulate vsum, optionally emit output.
// grid = (64 k, 32 b), block = 128 (= D).
// ---------------------------------------------------------------------------
__global__ __launch_bounds__(128)
void caps_squash(const float* __restrict__ s, float* __restrict__ vsum,
                 float* __restrict__ out, int write_out) {
    __shared__ float red[128];
    const int t = threadIdx.x;
    const int k = blockIdx.x;
    const int b = blockIdx.y;
    const size_t idx = ((size_t)b * N_CAP + k) * D_CAP + t;

    float sv = s[idx];
    red[t] = sv * sv;
    __syncthreads();
#pragma unroll
    for (int off = 64; off > 0; off >>= 1) {
        if (t < off) red[t] += red[t + off];
        __syncthreads();
    }
    float sq = red[0];
    float coef = (sq / (1.0f + sq)) * rsqrtf(sq + 1e-9f);
    float v = sv * coef;
    vsum[idx] += v;
    if (write_out) out[idx] = v;   // (B, N, D, 1) flat
}

// ---------------------------------------------------------------------------
extern "C" void kernel_launch(void* const* d_in, const int* in_sizes, int n_in,
                              void* d_out, int out_size, void* d_ws, size_t ws_size,
                              hipStream_t stream) {
    const float* x = (const float*)d_in[0];   // (32, 64, 128)
    const float* W = (const float*)d_in[1];   // (1, 64, 64, 128, 128)
    float* out = (float*)d_out;               // (32, 64, 128, 1)

    float* u    = (float*)d_ws;                          // 64 MB
    float* s    = u + (size_t)B_SZ * U_PER_B;            // 1 MB
    float* vsum = s + (size_t)B_SZ * N_CAP * D_CAP;      // 1 MB
    const size_t sv_bytes = (size_t)B_SZ * N_CAP * D_CAP * sizeof(float);

    hipMemsetAsync(vsum, 0, sv_bytes, stream);

    // u = W @ x   (64 MB written once, W streamed once)
    caps_gemm_wmma<<<dim3(128, 64), 128, 0, stream>>>(W, x, u);

    // routing iteration 0 (uniform c), 1, 2
    for (int it = 0; it < 3; ++it) {
        hipMemsetAsync(s, 0, sv_bytes, stream);
        caps_routing_step<<<dim3(8, 32), 128, 0, stream>>>(u, vsum, s, it == 0 ? 1 : 0);
        caps_squash<<<dim3(64, 32), 128, 0, stream>>>(s, vsum, out, it == 2 ? 1 : 0);
    }
}